// VectorQuantizer_cluster_4002909519937
// MI455X (gfx1250) — compile-verified
//
#include <hip/hip_runtime.h>
#include <hip/hip_bf16.h>
#include <cfloat>

#define N_ROWS 131072
#define DIM    64
#define KCODES 1024
#define BETA   0.25f
#define P_BLOCKS 64
#define HALFK  512

typedef float v2f __attribute__((ext_vector_type(2)));
typedef float v8f __attribute__((ext_vector_type(8)));

// CDNA5 async global->LDS copy (ASYNCcnt-tracked), with safe fallback.
// Host pass parses the fallback; device pass uses the async op if declared.
#if defined(__HIP_DEVICE_COMPILE__) && __has_builtin(__builtin_amdgcn_global_load_async_to_lds_b32)
#define HAS_ASYNC_LDS 1
#else
#define HAS_ASYNC_LDS 0
#endif

__device__ __forceinline__ void stage_one(const float* g, float* l) {
#if HAS_ASYNC_LDS
  typedef __attribute__((address_space(1))) int gas_i32;   // global (prints as __device__)
  typedef __attribute__((address_space(3))) int las_i32;   // LDS    (prints as __shared__)
  __builtin_amdgcn_global_load_async_to_lds_b32((gas_i32*)g, (las_i32*)l, 0, 0);
#else
  *l = *g;
#endif
}

__device__ __forceinline__ void async_wait_all() {
#if HAS_ASYNC_LDS
  asm volatile("s_wait_asynccnt 0x0" ::: "memory");
#endif
}

// ---------------------------------------------------------------------------
// Kernel 1: argmin over codes via f32 WMMA GEMM, quantized gather, loss partial
// ---------------------------------------------------------------------------
__global__ __launch_bounds__(256) void vq_argmin_kernel(
    const float* __restrict__ x, const float* __restrict__ emb,
    float* __restrict__ out_q, int* __restrict__ idx_out,
    float* __restrict__ loss_part)
{
  __shared__ float s_enorm[KCODES];      // ||e_k||^2
  __shared__ float s_bt[2][16 * 66];     // double-buffered transposed E tile
  __shared__ float s_lossw[8];

  const int tid  = threadIdx.x;
  const int lane = tid & 31;
  const int wave = tid >> 5;
  const int m    = lane & 15;            // row within tile / code within tile
  const int h    = lane >> 4;            // half-wave

  // --- codebook column squared norms (cooperative, 4 codes per thread) ---
  {
    const int k4 = tid * 4;
    float ax = 0.f, ay = 0.f, az = 0.f, aw = 0.f;
    for (int d = 0; d < DIM; ++d) {
      float4 e = *(const float4*)(emb + (size_t)d * KCODES + k4);
      ax = fmaf(e.x, e.x, ax); ay = fmaf(e.y, e.y, ay);
      az = fmaf(e.z, e.z, az); aw = fmaf(e.w, e.w, aw);
    }
    s_enorm[k4 + 0] = ax; s_enorm[k4 + 1] = ay;
    s_enorm[k4 + 2] = az; s_enorm[k4 + 3] = aw;
  }

  // --- A fragments: 16 rows x 64 of x, ISA layout for WMMA_F32_16X16X4 ---
  // lane L: row = m, VGPR0 = K=(4c+2h), VGPR1 = K=(4c+2h+1)  -> float2 loads
  const int rowBase = blockIdx.x * 128 + wave * 16;
  const float* xr = x + (size_t)(rowBase + m) * DIM + 2 * h;
  v2f a[16];
#pragma unroll
  for (int c = 0; c < 16; ++c) a[c] = *(const v2f*)(xr + 4 * c);

  const int sn  = tid & 15;              // staging: code within tile
  const int sd0 = tid >> 4;              // staging: base d

  // prologue: stage tile 0 into buffer 0 (transposed: s_bt[buf][code%16][d])
#pragma unroll
  for (int j = 0; j < 4; ++j) {
    const int d = sd0 + j * 16;
    stage_one(emb + (size_t)d * KCODES + sn, &s_bt[0][sn * 66 + d]);
  }
  async_wait_all();
  __syncthreads();

  float minv[8];
  int   mini[8];
#pragma unroll
  for (int j = 0; j < 8; ++j) { minv[j] = FLT_MAX; mini[j] = 0; }

  for (int kt = 0; kt < KCODES / 16; ++kt) {
    const int cb = kt * 16;
    const float* buf = s_bt[kt & 1];

    // stage NEXT tile into the other buffer; overlaps with the WMMAs below
    if (kt + 1 < KCODES / 16) {
#pragma unroll
      for (int j = 0; j < 4; ++j) {
        const int d = sd0 + j * 16;
        stage_one(emb + (size_t)d * KCODES + cb + 16 + sn,
                  &s_bt[(kt + 1) & 1][sn * 66 + d]);
        if (kt + 2 < KCODES / 16)
          __builtin_prefetch(emb + (size_t)d * KCODES + cb + 32 + sn, 0, 2);
      }
    }

    // sim tile [16 rows x 16 codes] = sum over 16 chained K=4 WMMAs
    v8f c = {};
#pragma unroll
    for (int q = 0; q < 16; ++q) {
      v2f b = *(const v2f*)(&buf[m * 66 + 4 * q + 2 * h]);  // B: K=(4q+2h),(4q+2h+1), N=m
      c = __builtin_amdgcn_wmma_f32_16x16x4_f32(
          /*neg_a=*/false, a[q], /*neg_b=*/false, b,
          /*c_mod=*/(short)0, c, /*reuse_a=*/false, /*reuse_b=*/false);
    }

    // s_k = ||e_k||^2 - 2 * sim  (||x||^2 is row-constant, irrelevant for argmin)
    const float en   = s_enorm[cb + m];
    const int   code = cb + m;
#pragma unroll
    for (int j = 0; j < 8; ++j) {
      const float s = fmaf(-2.0f, c[j], en);
      if (s < minv[j]) { minv[j] = s; mini[j] = code; }
    }

    async_wait_all();   // next tile fully landed in LDS
    __syncthreads();    // all waves done reading current buffer / writes visible
  }

  // --- cross-lane argmin within each 16-lane group (tie -> lowest code) ---
#pragma unroll
  for (int j = 0; j < 8; ++j) {
    float v = minv[j]; int ix = mini[j];
#pragma unroll
    for (int off = 8; off >= 1; off >>= 1) {
      const float ov = __shfl_xor(v, off, 32);
      const int   oi = __shfl_xor(ix, off, 32);
      if (ov < v || (ov == v && oi < ix)) { v = ov; ix = oi; }
    }
    minv[j] = v; mini[j] = ix;
  }
  if ((lane & 15) == 0) {
#pragma unroll
    for (int j = 0; j < 8; ++j) idx_out[rowBase + 8 * h + j] = mini[j];
  }

  // --- quantized gather + commitment-loss partial ---
  float lacc = 0.f;
#pragma unroll 4
  for (int r = 0; r < 16; ++r) {
    const int code = __shfl(mini[r & 7], (r >> 3) << 4, 32);
    const size_t ro = (size_t)(rowBase + r) * DIM;
    const float e0 = emb[(size_t)lane * KCODES + code];
    const float e1 = emb[(size_t)(lane + 32) * KCODES + code];
    const float x0 = x[ro + lane];
    const float x1 = x[ro + lane + 32];
    out_q[ro + lane]      = e0;
    out_q[ro + lane + 32] = e1;
    const float d0 = e0 - x0, d1 = e1 - x1;
    lacc = fmaf(d0, d0, lacc);
    lacc = fmaf(d1, d1, lacc);
  }
#pragma unroll
  for (int off = 16; off >= 1; off >>= 1) lacc += __shfl_xor(lacc, off, 32);
  if (lane == 0) s_lossw[wave] = lacc;
  __syncthreads();
  if (tid == 0) {
    float t = 0.f;
    for (int w = 0; w < 8; ++w) t += s_lossw[w];
    loss_part[blockIdx.x] = t;
  }
}

// ---------------------------------------------------------------------------
// Kernel 2: segment-sum via LDS privatization (ds_add_f32), two K-half passes
// ---------------------------------------------------------------------------
__global__ __launch_bounds__(256) void vq_scatter_kernel(
    const float* __restrict__ x, const int* __restrict__ idx,
    float* __restrict__ psum, float* __restrict__ pcnt)
{
  extern __shared__ float smem[];          // HALFK*DIM sums + HALFK counts
  float* ssum = smem;
  float* scnt = smem + HALFK * DIM;

  const int tid  = threadIdx.x;
  const int lane = tid & 31;
  const int wave = tid >> 5;
  const int rows_per_block = N_ROWS / P_BLOCKS;   // 2048
  const int row0 = blockIdx.x * rows_per_block;

  for (int pass = 0; pass < 2; ++pass) {
    const int kbase = pass * HALFK;
    for (int i = tid; i < HALFK * DIM + HALFK; i += 256) smem[i] = 0.f;
    __syncthreads();

    for (int r = wave; r < rows_per_block; r += 8) {
      const int row  = row0 + r;
      const int code = idx[row];
      const int lk   = code - kbase;
      if ((unsigned)lk < (unsigned)HALFK) {
        const size_t ro = (size_t)row * DIM;
        const float v0 = x[ro + lane];
        const float v1 = x[ro + lane + 32];
        atomicAdd(&ssum[lk * DIM + lane], v0);
        atomicAdd(&ssum[lk * DIM + lane + 32], v1);
        if (lane == 0) atomicAdd(&scnt[lk], 1.0f);
      }
    }
    __syncthreads();

    float* ps = psum + (size_t)blockIdx.x * (KCODES * DIM) + (size_t)kbase * DIM;
    for (int i = tid; i < HALFK * DIM; i += 256) ps[i] = ssum[i];
    float* pc = pcnt + (size_t)blockIdx.x * KCODES + kbase;
    for (int i = tid; i < HALFK; i += 256) pc[i] = scnt[i];
    __syncthreads();
  }
}

// ---------------------------------------------------------------------------
// Kernel 3: reduce partials, finalize counts / emb_sum / embeddings / loss
// ---------------------------------------------------------------------------
__global__ __launch_bounds__(256) void vq_finalize_kernel(
    const float* __restrict__ sample_count, const float* __restrict__ emb_sum_in,
    const float* __restrict__ psum, const float* __restrict__ pcnt,
    const float* __restrict__ loss_part,
    float* __restrict__ out_loss, float* __restrict__ out_emb,
    float* __restrict__ out_cnt, float* __restrict__ out_sum)
{
  const int k = blockIdx.x * 256 + threadIdx.x;   // one code column per thread

  float cnt = sample_count[k];
  for (int p = 0; p < P_BLOCKS; ++p) cnt += pcnt[(size_t)p * KCODES + k];

  float acc[DIM];
#pragma unroll
  for (int d = 0; d < DIM; ++d) acc[d] = 0.f;
  for (int p = 0; p < P_BLOCKS; ++p) {
    const float4* ps = (const float4*)(psum + (size_t)p * KCODES * DIM + (size_t)k * DIM);
#pragma unroll
    for (int d4 = 0; d4 < DIM / 4; ++d4) {
      const float4 v = ps[d4];
      acc[4 * d4 + 0] += v.x; acc[4 * d4 + 1] += v.y;
      acc[4 * d4 + 2] += v.z; acc[4 * d4 + 3] += v.w;
    }
  }

  const float inv = 1.0f / fmaxf(cnt, 1e-5f);
#pragma unroll
  for (int d = 0; d < DIM; ++d) {
    const float s = emb_sum_in[(size_t)d * KCODES + k] + acc[d];
    out_sum[(size_t)d * KCODES + k] = s;
    out_emb[(size_t)d * KCODES + k] = s * inv;
  }
  out_cnt[k] = cnt;

  if (k == 0) {
    float l = 0.f;
    for (int b = 0; b < N_ROWS / 128; ++b) l += loss_part[b];
    out_loss[0] = BETA * l / (float)(N_ROWS * DIM);
  }
}

// ---------------------------------------------------------------------------
extern "C" void kernel_launch(void* const* d_in, const int* in_sizes, int n_in,
                              void* d_out, int out_size, void* d_ws, size_t ws_size,
                              hipStream_t stream)
{
  (void)in_sizes; (void)n_in; (void)out_size; (void)ws_size;
  const float* x      = (const float*)d_in[0];   // [N, D]
  const float* emb    = (const float*)d_in[1];   // [D, K]
  const float* scount = (const float*)d_in[2];   // [K]
  const float* esum   = (const float*)d_in[3];   // [D, K]

  float* out       = (float*)d_out;
  float* out_q     = out;                                   // N*D
  float* out_loss  = out + (size_t)N_ROWS * DIM;            // 1
  float* out_emb   = out_loss + 1;                          // D*K
  float* out_cnt   = out_emb + (size_t)DIM * KCODES;        // K
  float* out_sum   = out_cnt + KCODES;                      // D*K

  float* ws        = (float*)d_ws;
  float* loss_part = ws;                                    // N/128 = 1024
  int*   idx       = (int*)(ws + 1024);                     // N
  float* psum      = ws + 1024 + N_ROWS;                    // P*K*D
  float* pcnt      = psum + (size_t)P_BLOCKS * KCODES * DIM; // P*K

  vq_argmin_kernel<<<N_ROWS / 128, 256, 0, stream>>>(x, emb, out_q, idx, loss_part);

  const size_t smem2 = (size_t)(HALFK * DIM + HALFK) * sizeof(float); // 133,120 B
  (void)hipFuncSetAttribute(reinterpret_cast<const void*>(&vq_scatter_kernel),
                            hipFuncAttributeMaxDynamicSharedMemorySize, (int)smem2);
  vq_scatter_kernel<<<P_BLOCKS, 256, smem2, stream>>>(x, idx, psum, pcnt);

  vq_finalize_kernel<<<KCODES / 256, 256, 0, stream>>>(
      scount, esum, psum, pcnt, loss_part,
      out_loss, out_emb, out_cnt, out_sum);
}